// GraphTransformer_3212635537993
// MI455X (gfx1250) — compile-verified
//
#include <hip/hip_runtime.h>
#include <hip/hip_bf16.h>
#include <math.h>

// ---------------------------------------------------------------------------
// GAT layer for MI455X (gfx1250, wave32).
//   k_nodegemm : ngl = X@Wl^T, ngr = X@Wr^T  via V_WMMA_F32_16X16X4_F32
//   k_edge_score: per-edge attention logits -> exp -> atomic denom (no
//                 segment_max needed: softmax is shift-invariant and logits
//                 are O(1) here, exp cannot overflow)
//   k_edge_agg : out[src] += ngr[dst] * alpha   (L2 f32 atomics)
//   k_ffn      : fused [out|X]@W1^T + b1 -> exact GELU -> @W2^T + b2 (WMMA,
//                LDS re-layout between the two GEMMs)
// ---------------------------------------------------------------------------

#define NN   50000
#define EE   800000
#define D    64          // DIN == DEMB == DFFN == 64
#define HH   4
#define DH   16
#define SLOPE 0.2f

typedef float v2f __attribute__((ext_vector_type(2)));
typedef float v8f __attribute__((ext_vector_type(8)));

#define TILES ((NN + 15) / 16)     // 3125
#define WPB   8                    // waves per block (256 threads)

// --------------------------------------------------------------------------
// Kernel 1: node GEMMs.  One wave computes a 16x64 strip of ngl and ngr.
// A(16x4 f32) layout: lanes 0-15 -> M=lane, K = 4s+{0,1}; lanes 16-31 -> K+2.
// B(4x16) layout: lanes 0-15 -> N=lane, K = 4s+{0,1}; lanes 16-31 -> K+2.
// C(16x16): VGPR r holds M=r (lanes 0-15) / M=r+8 (lanes 16-31), N=lane%16.
// --------------------------------------------------------------------------
__global__ void k_nodegemm(const float* __restrict__ attr,
                           const float* __restrict__ Wl,
                           const float* __restrict__ Wr,
                           float* __restrict__ ngl,
                           float* __restrict__ ngr) {
  const int wave = threadIdx.x >> 5;
  const int lane = threadIdx.x & 31;
  const int lr   = lane & 15;
  const int hi   = lane >> 4;
  const int tile = blockIdx.x * WPB + wave;
  const int m0   = tile * 16;

  int arow = m0 + lr;
  if (arow >= NN) arow = NN - 1;               // clamp; stores are guarded
  const float* __restrict__ ap = attr + (size_t)arow * D;

  v8f accL[4] = {v8f{}, v8f{}, v8f{}, v8f{}};
  v8f accR[4] = {v8f{}, v8f{}, v8f{}, v8f{}};

#pragma unroll
  for (int s = 0; s < 16; ++s) {
    const int k = 4 * s + 2 * hi;
    v2f a;
    a.x = ap[k];
    a.y = ap[k + 1];
#pragma unroll
    for (int nt = 0; nt < 4; ++nt) {
      const int n = nt * 16 + lr;
      v2f bl, br;
      bl.x = Wl[n * D + k];     bl.y = Wl[n * D + k + 1];
      br.x = Wr[n * D + k];     br.y = Wr[n * D + k + 1];
      accL[nt] = __builtin_amdgcn_wmma_f32_16x16x4_f32(
          false, a, false, bl, (short)0, accL[nt], false, false);
      accR[nt] = __builtin_amdgcn_wmma_f32_16x16x4_f32(
          false, a, false, br, (short)0, accR[nt], false, false);
    }
  }

#pragma unroll
  for (int r = 0; r < 8; ++r) {
    const int m = m0 + r + 8 * hi;
    if (m < NN) {
#pragma unroll
      for (int nt = 0; nt < 4; ++nt) {
        ngl[(size_t)m * D + nt * 16 + lr] = accL[nt][r];
        ngr[(size_t)m * D + nt * 16 + lr] = accR[nt][r];
      }
    }
  }
}

// --------------------------------------------------------------------------
// Kernel 2: per-directed-edge attention logits.  One wave per edge.
// Lane l owns dims 2l,2l+1; head = l>>3; 8-lane shfl-xor tree reduction.
// --------------------------------------------------------------------------
__global__ void k_edge_score(const int* __restrict__ ei,
                             const float* __restrict__ ngl,
                             const float* __restrict__ ngr,
                             const float* __restrict__ att,
                             float* __restrict__ exbuf,
                             float* __restrict__ denom) {
  const int wave = threadIdx.x >> 5;
  const int lane = threadIdx.x & 31;
  const long e = (long)blockIdx.x * WPB + wave;
  if (e >= 2L * EE) return;

  int src, dst;
  if (e < EE) { src = ei[2 * e];           dst = ei[2 * e + 1]; }
  else        { src = ei[2 * (e - EE) + 1]; dst = ei[2 * (e - EE)]; }

  const int d0 = lane * 2;
  const float2 gl = *(const float2*)(ngl + (size_t)src * D + d0);
  const float2 gr = *(const float2*)(ngr + (size_t)dst * D + d0);
  float v0 = gl.x + gr.x, v1 = gl.y + gr.y;
  v0 = v0 > 0.f ? v0 : SLOPE * v0;
  v1 = v1 > 0.f ? v1 : SLOPE * v1;
  float t = att[d0] * v0 + att[d0 + 1] * v1;
  t += __shfl_xor(t, 1);
  t += __shfl_xor(t, 2);
  t += __shfl_xor(t, 4);

  if ((lane & 7) == 0) {
    const int h = lane >> 3;
    const float ex = expf(t);
    exbuf[e * HH + h] = ex;
    atomicAdd(&denom[(size_t)src * HH + h], ex);
  }
}

// --------------------------------------------------------------------------
// Kernel 3: aggregate  out[src] += ngr[dst] * alpha.  One wave per edge.
// --------------------------------------------------------------------------
__global__ void k_edge_agg(const int* __restrict__ ei,
                           const float* __restrict__ ngr,
                           const float* __restrict__ exbuf,
                           const float* __restrict__ denom,
                           float* __restrict__ outacc) {
  const int wave = threadIdx.x >> 5;
  const int lane = threadIdx.x & 31;
  const long e = (long)blockIdx.x * WPB + wave;
  if (e >= 2L * EE) return;

  int src, dst;
  if (e < EE) { src = ei[2 * e];           dst = ei[2 * e + 1]; }
  else        { src = ei[2 * (e - EE) + 1]; dst = ei[2 * (e - EE)]; }

  const int d0 = lane * 2;
  const int h  = lane >> 3;
  const float alpha =
      exbuf[e * HH + h] / (denom[(size_t)src * HH + h] + 1e-16f);
  const float2 g = *(const float2*)(ngr + (size_t)dst * D + d0);
  atomicAdd(&outacc[(size_t)src * D + d0],     g.x * alpha);
  atomicAdd(&outacc[(size_t)src * D + d0 + 1], g.y * alpha);
}

// --------------------------------------------------------------------------
// Kernel 4: fused FFN. A = [outacc | node_attr] (K=128) @ W1^T -> GELU,
// re-layout C->A via wave-private LDS, then @ W2^T + b2 -> d_out.
// LDS row stride 68 floats keeps the two 16-lane halves on disjoint banks.
// --------------------------------------------------------------------------
#define LSTRIDE 68
__global__ void k_ffn(const float* __restrict__ outacc,
                      const float* __restrict__ attr,
                      const float* __restrict__ W1,
                      const float* __restrict__ b1,
                      const float* __restrict__ W2,
                      const float* __restrict__ b2,
                      float* __restrict__ out) {
  __shared__ float lds[WPB * 16 * LSTRIDE];

  const int wave = threadIdx.x >> 5;
  const int lane = threadIdx.x & 31;
  const int lr   = lane & 15;
  const int hi   = lane >> 4;
  const int tile = blockIdx.x * WPB + wave;
  const int m0   = tile * 16;

  int arow = m0 + lr;
  if (arow >= NN) arow = NN - 1;
  const float* __restrict__ accp = outacc + (size_t)arow * D;
  const float* __restrict__ attp = attr   + (size_t)arow * D;

  // ---- GEMM 1: (16x128) @ (128x64) ----
  v8f z[4] = {v8f{}, v8f{}, v8f{}, v8f{}};
#pragma unroll
  for (int s = 0; s < 32; ++s) {
    const int k = 4 * s + 2 * hi;
    v2f a;
    if (k < D) { a.x = accp[k];     a.y = accp[k + 1]; }
    else       { a.x = attp[k - D]; a.y = attp[k - D + 1]; }
#pragma unroll
    for (int nt = 0; nt < 4; ++nt) {
      const int n = nt * 16 + lr;
      v2f b;
      b.x = W1[n * 128 + k];
      b.y = W1[n * 128 + k + 1];
      z[nt] = __builtin_amdgcn_wmma_f32_16x16x4_f32(
          false, a, false, b, (short)0, z[nt], false, false);
    }
  }

  // ---- bias + exact GELU, stage to LDS in row-major tile form ----
  float* __restrict__ my = &lds[wave * 16 * LSTRIDE];
#pragma unroll
  for (int nt = 0; nt < 4; ++nt) {
    const float bb = b1[nt * 16 + lr];
#pragma unroll
    for (int r = 0; r < 8; ++r) {
      const float x = z[nt][r] + bb;
      const float g = 0.5f * x * (1.0f + erff(x * 0.70710678118654752f));
      my[(r + 8 * hi) * LSTRIDE + nt * 16 + lr] = g;
    }
  }
  __syncthreads();   // uniform: every wave reaches here

  // ---- GEMM 2: (16x64) @ (64x64) ----
  v8f y[4] = {v8f{}, v8f{}, v8f{}, v8f{}};
#pragma unroll
  for (int s = 0; s < 16; ++s) {
    const int k = 4 * s + 2 * hi;
    v2f a;
    a.x = my[lr * LSTRIDE + k];
    a.y = my[lr * LSTRIDE + k + 1];
#pragma unroll
    for (int nt = 0; nt < 4; ++nt) {
      const int n = nt * 16 + lr;
      v2f b;
      b.x = W2[n * D + k];
      b.y = W2[n * D + k + 1];
      y[nt] = __builtin_amdgcn_wmma_f32_16x16x4_f32(
          false, a, false, b, (short)0, y[nt], false, false);
    }
  }

#pragma unroll
  for (int nt = 0; nt < 4; ++nt) {
    const float bb = b2[nt * 16 + lr];
#pragma unroll
    for (int r = 0; r < 8; ++r) {
      const int m = m0 + r + 8 * hi;
      if (m < NN) out[(size_t)m * D + nt * 16 + lr] = y[nt][r] + bb;
    }
  }
}

// --------------------------------------------------------------------------
extern "C" void kernel_launch(void* const* d_in, const int* in_sizes, int n_in,
                              void* d_out, int out_size, void* d_ws, size_t ws_size,
                              hipStream_t stream) {
  const float* attr = (const float*)d_in[0];   // (N,64)
  const int*   ei   = (const int*)  d_in[1];   // (E,2)
  const float* Wl   = (const float*)d_in[2];   // (64,64)
  const float* Wr   = (const float*)d_in[3];   // (64,64)
  const float* att  = (const float*)d_in[4];   // (1,4,16)
  const float* W1   = (const float*)d_in[5];   // (64,128)
  const float* b1   = (const float*)d_in[6];   // (64,)
  const float* W2   = (const float*)d_in[7];   // (64,64)
  const float* b2   = (const float*)d_in[8];   // (64,)
  float* outp = (float*)d_out;                 // (N,64)

  // workspace layout (floats)
  float* ngl    = (float*)d_ws;                 // N*64
  float* ngr    = ngl    + (size_t)NN * D;      // N*64
  float* outacc = ngr    + (size_t)NN * D;      // N*64   (must be zeroed)
  float* denom  = outacc + (size_t)NN * D;      // N*4    (must be zeroed)
  float* exbuf  = denom  + (size_t)NN * HH;     // 2E*4

  // zero the accumulators (outacc and denom are contiguous)
  hipMemsetAsync(outacc, 0, ((size_t)NN * D + (size_t)NN * HH) * sizeof(float),
                 stream);

  const int gemmBlocks = (TILES + WPB - 1) / WPB;          // 391
  const long nEdges    = 2L * EE;                          // 1.6M
  const int edgeBlocks = (int)((nEdges + WPB - 1) / WPB);  // 200000

  k_nodegemm<<<gemmBlocks, 256, 0, stream>>>(attr, Wl, Wr, ngl, ngr);
  k_edge_score<<<edgeBlocks, 256, 0, stream>>>(ei, ngl, ngr, att, exbuf, denom);
  k_edge_agg<<<edgeBlocks, 256, 0, stream>>>(ei, ngr, exbuf, denom, outacc);
  k_ffn<<<gemmBlocks, 256, 0, stream>>>(outacc, attr, W1, b1, W2, b2, outp);
}